// Yolov3_7636451852904
// MI455X (gfx1250) — compile-verified
//
#include <hip/hip_runtime.h>

// ---------------------------------------------------------------------------
// YOLOv3 NMS post-processing for MI455X (gfx1250, wave32)
//   B=16 batches, N=10647 predictions, 85 features (4 box, 1 obj, 80 cls)
//   PRE_NMS=2048, MAX_DET=300, CONF=0.5, NMS=0.5, MAX_WH=4096, scale=1.0
// ---------------------------------------------------------------------------

#define NBATCH   16
#define NPRED    10647
#define NFEAT    85
#define NCLS     80
#define PRE_NMS  2048
#define MAX_DET  300
#define CONF_TH  0.5f
#define NMS_TH   0.5f

// per-batch workspace layout (byte offsets, all 256B aligned)
#define S_SCORE   0u         // float[N]           42588 B
#define S_BOX     44032u     // float4[N]         170352 B (raw xyxy, no class offset)
#define S_CLS     215040u    // int[N]             42588 B
#define S_TSCORE  258048u    // float[2048]
#define S_TBOX    266240u    // float4[2048]       (raw xyxy)
#define S_TCLS    299008u    // int[2048]
#define S_MASK    307200u    // u32[2048][64]     524288 B : bit j of mask[i][*] = (iou(i,j)>thr)
#define S_KEEP    831488u    // u32[64]
#define BATCH_STRIDE 835584u // 16 * 835584 = 13,369,344 B total workspace

typedef float v2f __attribute__((ext_vector_type(2)));
typedef float v8f __attribute__((ext_vector_type(8)));

__device__ __forceinline__ unsigned int fkey(float f) {
  // monotonic float -> uint mapping (handles negatives); bigger float -> bigger uint
  unsigned int x = __float_as_uint(f);
  return (x & 0x80000000u) ? ~x : (x | 0x80000000u);
}

// LDS byte offset of a __shared__ object: generic pointers to LDS carry the
// workgroup-relative offset in their low 32 bits on AMDGPU.
#define LDS_OFF(p) ((unsigned int)(size_t)(p))

// ---------------------------------------------------------------------------
// K1: per-prediction score / class argmax / xywh->xyxy.  One wave32 per row.
// ---------------------------------------------------------------------------
__global__ void k_score(const float* __restrict__ pred, char* __restrict__ ws,
                        int totalPred) {
  int wid  = blockIdx.x * (blockDim.x >> 5) + (threadIdx.x >> 5);
  int lane = threadIdx.x & 31;
  if (wid >= totalPred) return;
  const float* p = pred + (size_t)wid * NFEAT;

  // lanes cover classes {lane, lane+32, lane+64} (coalesced); stable argmax
  float v = p[5 + lane];
  int   c = lane;
  float t1 = p[5 + 32 + lane];
  if (t1 > v) { v = t1; c = lane + 32; }
  if (lane < 16) {
    float t2 = p[5 + 64 + lane];
    if (t2 > v) { v = t2; c = lane + 64; }
  }
  for (int off = 16; off >= 1; off >>= 1) {
    float ov = __shfl_down(v, off, 32);
    int   oc = __shfl_down(c, off, 32);
    if (ov > v || (ov == v && oc < c)) { v = ov; c = oc; }
  }
  if (lane == 0) {
    float x = p[0], y = p[1], w = p[2], h = p[3], obj = p[4];
    bool  valid = (obj >= CONF_TH);
    float score = valid ? (v * obj) : -1.0f;   // conf = obj * max(cls)
    int b = wid / NPRED, i = wid % NPRED;
    char* base = ws + (size_t)b * BATCH_STRIDE;
    ((float*)(base + S_SCORE))[i] = score;
    float4 bx;
    bx.x = x - 0.5f * w; bx.y = y - 0.5f * h;
    bx.z = x + 0.5f * w; bx.w = y + 0.5f * h;
    ((float4*)(base + S_BOX))[i] = bx;
    ((int*)(base + S_CLS))[i]    = c;
  }
}

// ---------------------------------------------------------------------------
// K2: per-batch full bitonic sort of 16384 padded keys, entirely in 128KB LDS
//     (CDNA5: 320KB LDS per WGP).  key = (~fkey(score))<<32 | idx  so ascending
//     u64 sort == stable descending score sort (ties -> lower idx first).
// ---------------------------------------------------------------------------
__global__ void k_sort_top(char* __restrict__ ws) {
  extern __shared__ unsigned long long sm[];   // 16384 * 8B = 128 KB dynamic LDS
  const int M = 16384;
  int b = blockIdx.x, tid = threadIdx.x, bd = blockDim.x;
  char* base = ws + (size_t)b * BATCH_STRIDE;
  const float* sc = (const float*)(base + S_SCORE);

  for (int i = tid; i < M; i += bd) {
    float s = (i < NPRED) ? sc[i] : -1.0e30f;    // pad sorts last (scores >= -1)
    sm[i] = ((unsigned long long)(~fkey(s)) << 32) | (unsigned int)i;
  }
  __syncthreads();
  for (int k = 2; k <= M; k <<= 1) {
    for (int j = k >> 1; j > 0; j >>= 1) {
      for (int i = tid; i < M; i += bd) {
        int pr = i ^ j;
        if (pr > i) {
          unsigned long long a = sm[i], bb = sm[pr];
          bool up = ((i & k) == 0);
          if ((a > bb) == up) { sm[i] = bb; sm[pr] = a; }
        }
      }
      __syncthreads();
    }
  }
  const float4* bxAll = (const float4*)(base + S_BOX);
  const int*    clAll = (const int*)(base + S_CLS);
  float*  ts = (float*)(base + S_TSCORE);
  float4* tb = (float4*)(base + S_TBOX);
  int*    tc = (int*)(base + S_TCLS);
  for (int k = tid; k < PRE_NMS; k += bd) {
    unsigned int idx = (unsigned int)sm[k];
    ts[k] = sc[idx];
    tb[k] = bxAll[idx];
    tc[k] = clAll[idx];
  }
}

// ---------------------------------------------------------------------------
// K3: build 2048x2048 IoU>thr bit matrix.  One wave per 16-column tile.
//
//  * Box/class tiles staged memory->LDS with GLOBAL_LOAD_ASYNC_TO_LDS_B128/B32
//    (gfx1250 async path, ASYNCcnt), no VGPR round-trip.
//  * Class-offset trick done exactly via class equality: boxes span < 4096, so
//    iou(offset boxes) > thr  <=>  (cls_i==cls_j) && iou(raw boxes) > thr.
//  * Pairwise area sums area[i]+area[j] via rank-2 outer-sum on the matrix
//    core:  D = [a_row, 1] x [[1],[a_col]]  with V_WMMA_F32_16X16X4_F32.
//    C/D layout: lane l, vgpr r  <->  (m = r + 8*(l>>4), n = l&15).
// ---------------------------------------------------------------------------
__global__ void k_iou_mask(char* __restrict__ ws) {
  int b    = blockIdx.x >> 7;       // 128 column tiles per batch
  int ci   = blockIdx.x & 127;
  int lane = threadIdx.x;           // block = 1 wave32
  int n    = lane & 15;
  int hi   = lane >> 4;

  char* base = ws + (size_t)b * BATCH_STRIDE;
  const float4* tb = (const float4*)(base + S_TBOX);
  const int*    tc = (const int*)(base + S_TCLS);
  unsigned int* mask = (unsigned int*)(base + S_MASK);

  __shared__ float4 colbox[16];
  __shared__ int    colcls[16];
  __shared__ float4 rowbox[16];
  __shared__ int    rowcls[16];

  // async-stage the 16 column boxes + classes into LDS
  if (lane < 16) {
    unsigned int ldsb = LDS_OFF(&colbox[lane]);
    unsigned int gob  = (unsigned int)((ci * 16 + lane) * sizeof(float4));
    asm volatile("global_load_async_to_lds_b128 %0, %1, %2 offset:0"
                 :: "v"(ldsb), "v"(gob), "s"(tb) : "memory");
    unsigned int ldsc = LDS_OFF(&colcls[lane]);
    unsigned int goc  = (unsigned int)((ci * 16 + lane) * sizeof(int));
    asm volatile("global_load_async_to_lds_b32 %0, %1, %2 offset:0"
                 :: "v"(ldsc), "v"(goc), "s"(tc) : "memory");
  }
  asm volatile("s_wait_asynccnt 0x0" ::: "memory");
  __syncthreads();

  float4 cb = colbox[n];
  int    cc = colcls[n];
  float carea = (cb.z - cb.x) * (cb.w - cb.y);

  // B (4x16): lanes 0-15 hold K=0,1 in vgpr 0,1; lanes 16-31 hold K=2,3 (zeros)
  v2f Bop; Bop[0] = hi ? 0.0f : 1.0f; Bop[1] = hi ? 0.0f : carea;

  for (int rj = 0; rj < 128; rj += 2) {
    unsigned int word = 0;
#pragma unroll
    for (int s = 0; s < 2; s++) {
      int jt = rj + s;
      // drain prior LDS reads of rowbox/rowcls before the DMA overwrites them
      asm volatile("s_wait_dscnt 0x0" ::: "memory");
      __syncthreads();
      if (lane < 16) {
        unsigned int ldsb = LDS_OFF(&rowbox[lane]);
        unsigned int gob  = (unsigned int)((jt * 16 + lane) * sizeof(float4));
        asm volatile("global_load_async_to_lds_b128 %0, %1, %2 offset:0"
                     :: "v"(ldsb), "v"(gob), "s"(tb) : "memory");
        unsigned int ldsc = LDS_OFF(&rowcls[lane]);
        unsigned int goc  = (unsigned int)((jt * 16 + lane) * sizeof(int));
        asm volatile("global_load_async_to_lds_b32 %0, %1, %2 offset:0"
                     :: "v"(ldsc), "v"(goc), "s"(tc) : "memory");
      }
      asm volatile("s_wait_asynccnt 0x0" ::: "memory");
      __syncthreads();

      // A (16x4): lane l holds row m=l&15; lanes 0-15 K=0,1; lanes 16-31 zeros
      float4 rn = rowbox[n];
      float ra = (rn.z - rn.x) * (rn.w - rn.y);
      v2f Aop; Aop[0] = hi ? 0.0f : ra; Aop[1] = hi ? 0.0f : 1.0f;
      v8f Cop = {};
      v8f D;
#if __has_builtin(__builtin_amdgcn_wmma_f32_16x16x4_f32)
      D = __builtin_amdgcn_wmma_f32_16x16x4_f32(
              false, Aop, false, Bop, (short)0, Cop, false, false);
#else
      // VALU fallback: pairwise area sums
#pragma unroll
      for (int r = 0; r < 8; r++) {
        float4 rb = rowbox[r + (hi << 3)];
        D[r] = (rb.z - rb.x) * (rb.w - rb.y) + carea;
      }
#endif
      unsigned int bits8 = 0;
#pragma unroll
      for (int r = 0; r < 8; r++) {
        int m = r + (hi << 3);
        float4 rb = rowbox[m];
        float lx = fmaxf(rb.x, cb.x), ly = fmaxf(rb.y, cb.y);
        float rx = fminf(rb.z, cb.z), ry = fminf(rb.w, cb.w);
        float iw = fmaxf(rx - lx, 0.0f), ih = fmaxf(ry - ly, 0.0f);
        float inter = iw * ih;
        float uni = D[r] - inter + 1e-16f;     // area_m + area_n - inter + eps
        bool hit = (rowcls[m] == cc) && (inter > NMS_TH * uni);  // iou>thr
        if (hit) bits8 |= (1u << r);
      }
      // assemble 16 row-bits per column: lane n has rows 0-7, lane n+16 rows 8-15
      unsigned int other = __shfl((int)bits8, lane + 16, 32);
      unsigned int chunk = (bits8 | (other << 8)) & 0xFFFFu;
      word |= chunk << (16 * s);
    }
    if (lane < 16) mask[(ci * 16 + n) * 64 + (rj >> 1)] = word;
  }
}

// ---------------------------------------------------------------------------
// K4: sequential greedy NMS scan.  One wave per batch; keep-mask lives in
//     registers (lane t owns words 2t,2t+1); next column prefetched into
//     registers, column after that prefetched into cache (global_prefetch_b8).
// ---------------------------------------------------------------------------
__global__ void k_nms(char* __restrict__ ws) {
  int b = blockIdx.x, lane = threadIdx.x;
  char* base = ws + (size_t)b * BATCH_STRIDE;
  const unsigned int* mask = (const unsigned int*)(base + S_MASK);
  const float* ts = (const float*)(base + S_TSCORE);
  unsigned int* keep = (unsigned int*)(base + S_KEEP);

  unsigned int kw0 = 0, kw1 = 0;
  uint2 m = ((const uint2*)mask)[lane];                 // column 0
  for (int i = 0; i < PRE_NMS; i++) {
    if (i + 2 < PRE_NMS)
      __builtin_prefetch((const void*)(mask + (size_t)(i + 2) * 64 + lane * 2), 0, 1);
    uint2 mn;
    if (i + 1 < PRE_NMS) mn = ((const uint2*)(mask + (size_t)(i + 1) * 64))[lane];
    else { mn.x = 0; mn.y = 0; }
    unsigned int v = (kw0 & m.x) | (kw1 & m.y);
    bool sup = __any(v != 0);
    float sc = ts[i];
    bool kp = (sc >= 0.0f) && !sup;                     // score==-1 <=> invalid
    if (kp && lane == (i >> 6)) {
      unsigned int bit = 1u << (i & 31);
      if (((i >> 5) & 1) == 0) kw0 |= bit; else kw1 |= bit;
    }
    m = mn;
  }
  keep[lane * 2]     = kw0;
  keep[lane * 2 + 1] = kw1;
}

// ---------------------------------------------------------------------------
// K5: final stable top-300 of (keep ? score : -1); writes all 16x300x6 outputs.
// ---------------------------------------------------------------------------
__global__ void k_final(char* __restrict__ ws, float* __restrict__ out) {
  __shared__ unsigned long long sm[PRE_NMS];            // 16 KB
  int b = blockIdx.x, tid = threadIdx.x, bd = blockDim.x;
  char* base = ws + (size_t)b * BATCH_STRIDE;
  const float*  ts = (const float*)(base + S_TSCORE);
  const float4* tb = (const float4*)(base + S_TBOX);
  const int*    tc = (const int*)(base + S_TCLS);
  const unsigned int* keep = (const unsigned int*)(base + S_KEEP);

  for (int i = tid; i < PRE_NMS; i += bd) {
    bool kp = (keep[i >> 5] >> (i & 31)) & 1u;
    float fs = kp ? ts[i] : -1.0f;
    sm[i] = ((unsigned long long)(~fkey(fs)) << 32) | (unsigned int)i;
  }
  __syncthreads();
  for (int k = 2; k <= PRE_NMS; k <<= 1) {
    for (int j = k >> 1; j > 0; j >>= 1) {
      for (int i = tid; i < PRE_NMS; i += bd) {
        int pr = i ^ j;
        if (pr > i) {
          unsigned long long a = sm[i], bb = sm[pr];
          bool up = ((i & k) == 0);
          if ((a > bb) == up) { sm[i] = bb; sm[pr] = a; }
        }
      }
      __syncthreads();
    }
  }
  for (int k = tid; k < MAX_DET; k += bd) {
    unsigned int idx = (unsigned int)sm[k];
    bool kp = (keep[idx >> 5] >> (idx & 31)) & 1u;
    float* o = out + ((size_t)b * MAX_DET + k) * 6;
    if (kp) {
      float4 bx = tb[idx];                              // scale = 416/416 = 1
      o[0] = bx.x; o[1] = bx.y; o[2] = bx.z; o[3] = bx.w;
      o[4] = ts[idx]; o[5] = (float)tc[idx];
    } else {
      o[0] = 0.f; o[1] = 0.f; o[2] = 0.f; o[3] = 0.f; o[4] = 0.f; o[5] = 0.f;
    }
  }
}

// ---------------------------------------------------------------------------
extern "C" void kernel_launch(void* const* d_in, const int* in_sizes, int n_in,
                              void* d_out, int out_size, void* d_ws, size_t ws_size,
                              hipStream_t stream) {
  const float* pred = (const float*)d_in[0];           // (16, 10647, 85) f32
  float* out = (float*)d_out;                          // (16, 300, 6)    f32
  char* ws = (char*)d_ws;                              // needs ~12.8 MB

  const int totalPred = NBATCH * NPRED;
  dim3 g1((totalPred + 7) / 8);                        // 8 waves / 256-thread block
  k_score<<<g1, 256, 0, stream>>>(pred, ws, totalPred);

  // 128 KB dynamic LDS per workgroup (CDNA5 WGP has 320 KB)
  k_sort_top<<<NBATCH, 1024, 16384 * sizeof(unsigned long long), stream>>>(ws);

  k_iou_mask<<<NBATCH * 128, 32, 0, stream>>>(ws);
  k_nms<<<NBATCH, 32, 0, stream>>>(ws);
  k_final<<<NBATCH, 256, 0, stream>>>(ws, out);
}